// GraphStack_48859547959298
// MI455X (gfx1250) — compile-verified
//
#include <hip/hip_runtime.h>
#include <hip/hip_bf16.h>

// ---------------------------------------------------------------------------
// GraphStack (3x GCNConv+GraphNorm, mean pool) for MI455X / gfx1250.
// Dense x@W runs on the f32 WMMA pipe (v_wmma_f32_16x16x4_f32, exact f32),
// with W staged in LDS in B-fragment order (ds_load_b64 per fragment).
// Edge aggregation = float4 gathers + non-returning global_atomic_add_f32.
// GraphNorm/pool exploit sorted `batch` via per-graph contiguous block reduces.
// ---------------------------------------------------------------------------

#define GS_N       100000
#define GS_E       1200000
#define GS_D       64
#define GS_L       3
#define GS_G       128
#define GS_EPS     1e-5f

typedef float v2f __attribute__((ext_vector_type(2)));
typedef float v8f __attribute__((ext_vector_type(8)));

// ---------------- degree / dinv ----------------
__global__ void gs_deg_count(const int* __restrict__ dst, float* __restrict__ deg, int n_edges) {
    int e = blockIdx.x * blockDim.x + threadIdx.x;
    if (e < n_edges) unsafeAtomicAdd(&deg[dst[e]], 1.0f);
}

__global__ void gs_make_dinv(float* __restrict__ deg_io, int n) {
    int i = blockIdx.x * blockDim.x + threadIdx.x;
    if (i < n) deg_io[i] = rsqrtf(deg_io[i] + 1.0f);   // +1 self-loop
}

// ---------------- graph start offsets (batch is sorted) ----------------
__global__ void gs_graph_starts(const int* __restrict__ batch, int* __restrict__ starts,
                                int n_nodes, int n_graphs) {
    int i = blockIdx.x * blockDim.x + threadIdx.x;
    if (i >= n_nodes) return;
    int b  = batch[i];
    int bp = (i == 0) ? -1 : batch[i - 1];
    for (int g = bp + 1; g <= b; ++g) starts[g] = i;
    if (i == n_nodes - 1)
        for (int g = b + 1; g <= n_graphs; ++g) starts[g] = n_nodes;
}

// ---------------- WMMA GEMM: H[N,64] = X[N,64] @ W[64,64] ----------------
// One wave -> 16 rows x 64 cols (4 accumulator tiles). 8 waves per block.
// W (16 KB) staged once per block into LDS, pre-swizzled to B-fragment order
// so each lane's fragment is a contiguous float2 -> single ds_load_b64.
__global__ __launch_bounds__(256)
void gs_gemm_xw(const float* __restrict__ X, const float* __restrict__ W,
                float* __restrict__ H, int n_rows) {
    __shared__ float ldsW[GS_D * GS_D];   // 16 KB, fragment order

    // Stage: flat idx = ks*256 + t*64 + lane*2 + j  maps to  W[k][n] with
    // k = ks*4 + 2*(lane/16) + j,  n = t*16 + lane%16.
    for (int idx = threadIdx.x; idx < GS_D * GS_D; idx += 256) {
        int j    = idx & 1;
        int lane = (idx >> 1) & 31;
        int t    = (idx >> 6) & 3;
        int ks   = idx >> 8;                       // k-step 0..15
        int k    = ks * 4 + ((lane >> 4) << 1) + j;
        int n    = t * 16 + (lane & 15);
        ldsW[idx] = W[k * GS_D + n];
    }
    __syncthreads();

    const int lane = threadIdx.x & 31;
    const int wave = threadIdx.x >> 5;
    const int row0 = (blockIdx.x * 8 + wave) * 16;
    if (row0 >= n_rows) return;                    // wave-uniform: EXEC all-1s

    const int mrow  = lane & 15;                   // M (and N for B/C frags)
    const int khalf = (lane >> 4) * 2;             // K sub-offset: 0 or 2

    v8f acc[4];
    #pragma unroll
    for (int t = 0; t < 4; ++t)
        #pragma unroll
        for (int v = 0; v < 8; ++v) acc[t][v] = 0.0f;

    // Branch-free row clamp: A row m only feeds D row m, and out-of-range D
    // rows are dropped at the store guard, so clamping is exact.
    const int row = min(row0 + mrow, n_rows - 1);
    const float* __restrict__ xrow  = X + (size_t)row * GS_D;
    const float* __restrict__ wfrag = ldsW + lane * 2;

    #pragma unroll
    for (int ks = 0; ks < 16; ++ks) {
        v2f a = *(const v2f*)(xrow + ks * 4 + khalf);   // global_load_b64
        #pragma unroll
        for (int t = 0; t < 4; ++t) {
            v2f b = *(const v2f*)(wfrag + (ks * 4 + t) * GS_D);  // ds_load_b64
            acc[t] = __builtin_amdgcn_wmma_f32_16x16x4_f32(
                false, a, false, b, (short)0, acc[t], false, false);
        }
    }

    // C/D layout: lane l, vgpr v -> (m = v + 8*(l/16), n = l%16);
    // lanes 0..15 write contiguous 64B rows -> coalesced b32 stores.
    const int mst = 8 * (lane >> 4);
    #pragma unroll
    for (int t = 0; t < 4; ++t)
        #pragma unroll
        for (int v = 0; v < 8; ++v) {
            int r = row0 + mst + v;
            if (r < n_rows) H[(size_t)r * GS_D + t * 16 + mrow] = acc[t][v];
        }
}

// ---------------- edge scatter: agg[dst] += h[src] * dinv[src]*dinv[dst] ----------------
// 16 threads per edge, float4 per thread; non-returning f32 atomics.
__global__ __launch_bounds__(256)
void gs_edge_scatter(const float* __restrict__ H, const int* __restrict__ src,
                     const int* __restrict__ dst, const float* __restrict__ dinv,
                     float* __restrict__ agg, int n_edges) {
    int t = blockIdx.x * blockDim.x + threadIdx.x;
    int e = t >> 4;
    if (e >= n_edges) return;
    int d4 = (t & 15) * 4;
    int s  = src[e];
    int d  = dst[e];
    float c = dinv[s] * dinv[d];
    const float4 hv = *reinterpret_cast<const float4*>(H + (size_t)s * GS_D + d4);
    float* out = agg + (size_t)d * GS_D + d4;
    unsafeAtomicAdd(out + 0, hv.x * c);
    unsafeAtomicAdd(out + 1, hv.y * c);
    unsafeAtomicAdd(out + 2, hv.z * c);
    unsafeAtomicAdd(out + 3, hv.w * c);
}

// ---------------- combine: y = agg + h*dinv^2 + bias (in-place on agg) ----------------
__global__ void gs_combine(float* __restrict__ y, const float* __restrict__ H,
                           const float* __restrict__ dinv, const float* __restrict__ bias,
                           int n_nodes) {
    int idx = blockIdx.x * blockDim.x + threadIdx.x;
    if (idx >= n_nodes * GS_D) return;
    int i = idx >> 6;   // node
    int d = idx & 63;   // feature
    float di = dinv[i];
    y[idx] = y[idx] + H[idx] * di * di + bias[d];
}

// ---------------- per-graph mean (also used for final mean-pool) ----------------
// One block per graph; 256 threads = 4 row-lanes x 64 dims.
__global__ __launch_bounds__(256)
void gs_seg_mean(const float* __restrict__ y, const int* __restrict__ starts,
                 float* __restrict__ mean_out) {
    int g = blockIdx.x;
    int s = starts[g], e = starts[g + 1];
    int d = threadIdx.x & 63;
    int r = threadIdx.x >> 6;
    float sum = 0.0f;
    for (int i = s + r; i < e; i += 4) sum += y[(size_t)i * GS_D + d];
    __shared__ float red[256];
    red[threadIdx.x] = sum;
    __syncthreads();
    if (r == 0) {
        float tot = red[d] + red[64 + d] + red[128 + d] + red[192 + d];
        float cnt = fmaxf((float)(e - s), 1.0f);
        mean_out[g * GS_D + d] = tot / cnt;
    }
}

// ---------------- GraphNorm: center (alpha*mean), var, scale ----------------
__global__ __launch_bounds__(256)
void gs_gn_norm(float* __restrict__ y, const int* __restrict__ starts,
                const float* __restrict__ mean, const float* __restrict__ alpha,
                const float* __restrict__ gamma, const float* __restrict__ beta) {
    int g = blockIdx.x;
    int s = starts[g], e = starts[g + 1];
    int d = threadIdx.x & 63;
    int r = threadIdx.x >> 6;
    float m  = mean[g * GS_D + d] * alpha[d];
    float sq = 0.0f;
    for (int i = s + r; i < e; i += 4) {
        size_t o = (size_t)i * GS_D + d;
        float c = y[o] - m;
        y[o] = c;
        sq += c * c;
    }
    __shared__ float red[256];
    __shared__ float rstd_s[64];
    red[threadIdx.x] = sq;
    __syncthreads();
    if (r == 0) {
        float var = (red[d] + red[64 + d] + red[128 + d] + red[192 + d]) /
                    fmaxf((float)(e - s), 1.0f);
        rstd_s[d] = rsqrtf(var + GS_EPS);
    }
    __syncthreads();
    float ga = gamma[d], be = beta[d], rstd = rstd_s[d];
    for (int i = s + r; i < e; i += 4) {
        size_t o = (size_t)i * GS_D + d;
        y[o] = ga * y[o] * rstd + be;
    }
}

// ---------------------------------------------------------------------------
extern "C" void kernel_launch(void* const* d_in, const int* in_sizes, int n_in,
                              void* d_out, int out_size, void* d_ws, size_t ws_size,
                              hipStream_t stream) {
    const float* x      = (const float*)d_in[0];
    const float* Ws     = (const float*)d_in[1];   // [3,64,64]
    const float* bs     = (const float*)d_in[2];   // [3,64]
    const float* gammas = (const float*)d_in[3];
    const float* betas  = (const float*)d_in[4];
    const float* alphas = (const float*)d_in[5];
    const int*   eidx   = (const int*)d_in[6];     // [2,E]
    const int*   batch  = (const int*)d_in[7];     // [N], sorted

    const int N = GS_N, E = GS_E, G = GS_G;
    const int* srcIdx = eidx;
    const int* dstIdx = eidx + E;

    // ---- workspace carve-out ----
    char*  ws  = (char*)d_ws;
    size_t off = 0;
    auto carve = [&](size_t bytes) -> void* {
        void* p = ws + off;
        off += (bytes + 255) & ~(size_t)255;
        return p;
    };
    float* dinv   = (float*)carve((size_t)N * 4);
    int*   starts = (int*)carve((size_t)(G + 1) * 4);
    float* mean   = (float*)carve((size_t)G * GS_D * 4);
    float* bufA   = (float*)carve((size_t)N * GS_D * 4);
    float* bufB   = (float*)carve((size_t)N * GS_D * 4);
    float* bufH   = (float*)carve((size_t)N * GS_D * 4);

    const int TB = 256;

    // ---- once per call: degrees, dinv, graph boundaries ----
    hipMemsetAsync(dinv, 0, (size_t)N * 4, stream);
    gs_deg_count<<<(E + TB - 1) / TB, TB, 0, stream>>>(dstIdx, dinv, E);
    gs_make_dinv<<<(N + TB - 1) / TB, TB, 0, stream>>>(dinv, N);
    gs_graph_starts<<<(N + TB - 1) / TB, TB, 0, stream>>>(batch, starts, N, G);

    // ---- 3 layers: GEMM (WMMA) -> scatter -> combine -> GraphNorm ----
    const int gemmBlocks    = (N + 127) / 128;                 // 128 rows / block
    const int scatterBlocks = (E * 16 + TB - 1) / TB;
    const int elemBlocks    = (N * GS_D + TB - 1) / TB;

    for (int l = 0; l < GS_L; ++l) {
        const float* cur  = (l == 0) ? x : ((l == 1) ? bufA : bufB);
        float*       outb = (l == 1) ? bufB : bufA;            // A, B, A

        gs_gemm_xw<<<gemmBlocks, TB, 0, stream>>>(cur, Ws + l * GS_D * GS_D, bufH, N);
        hipMemsetAsync(outb, 0, (size_t)N * GS_D * 4, stream);
        gs_edge_scatter<<<scatterBlocks, TB, 0, stream>>>(bufH, srcIdx, dstIdx, dinv, outb, E);
        gs_combine<<<elemBlocks, TB, 0, stream>>>(outb, bufH, dinv, bs + l * GS_D, N);
        gs_seg_mean<<<G, TB, 0, stream>>>(outb, starts, mean);
        gs_gn_norm<<<G, TB, 0, stream>>>(outb, starts, mean,
                                         alphas + l * GS_D, gammas + l * GS_D, betas + l * GS_D);
    }

    // ---- final global mean pool -> d_out [128, 64] ----
    gs_seg_mean<<<G, TB, 0, stream>>>(bufA, starts, (float*)d_out);
}